// LSTMAttentionJKNETGAT_29987461661047
// MI455X (gfx1250) — compile-verified
//
#include <hip/hip_runtime.h>
#include <cstdint>

// ---------------------------------------------------------------------------
// LSTMAttentionJKNETGAT for MI455X (gfx1250, wave32, WMMA)
//   - GAT GEMMs + all LSTM gate GEMMs on V_WMMA_F32_16X16X4_F32 (exact fp32)
//   - LDS tile staging via GLOBAL_LOAD_ASYNC_TO_LDS_B64/B128 (+ s_wait_asynccnt)
//   - Edge softmax aggregation via f32 atomics (acc fits in 192MB L2)
// ---------------------------------------------------------------------------

typedef float v2f __attribute__((ext_vector_type(2)));
typedef float v8f __attribute__((ext_vector_type(8)));

#define D_HID 256
#define D_IN  128

// ---------------------------------------------------------------------------
// helpers
// ---------------------------------------------------------------------------
__device__ __forceinline__ float sigm(float x) { return 1.0f / (1.0f + expf(-x)); }

__device__ __forceinline__ void atomicMaxF(float* addr, float val) {
  unsigned int* ua = (unsigned int*)addr;
  unsigned int old = *ua;
  while (__uint_as_float(old) < val) {
    unsigned int assumed = old;
    old = atomicCAS(ua, assumed, __float_as_uint(val));
    if (old == assumed) break;
  }
}

// CDNA5 async global->LDS copies (ASYNCcnt). LDS operand is the low 32 bits of
// the generic shared-space address (hardware uses addr[31:0] within LDS space).
__device__ __forceinline__ void async_ld_b64(const float* g, const float* l) {
  unsigned lo = (unsigned)(unsigned long long)(const void*)l;
  asm volatile("global_load_async_to_lds_b64 %0, %1, off" :: "v"(lo), "v"(g) : "memory");
}
__device__ __forceinline__ void async_ld_b128(const float* g, const float* l) {
  unsigned lo = (unsigned)(unsigned long long)(const void*)l;
  asm volatile("global_load_async_to_lds_b128 %0, %1, off" :: "v"(lo), "v"(g) : "memory");
}
__device__ __forceinline__ void wait_async0() {
  asm volatile("s_wait_asynccnt 0x0" ::: "memory");
}

__global__ void fill_kernel(float* __restrict__ p, float v, int n) {
  int i = blockIdx.x * blockDim.x + threadIdx.x;
  if (i < n) p[i] = v;
}

// ---------------------------------------------------------------------------
// Tiled fp32 WMMA GEMM: C[M,Nout] = A[M,K] @ B (+C if ACC) (+bias0+bias1)
//   BT=false: B is [K,Nout] row-major       -> LDS layout Bs[16][132]
//   BT=true : B is [Nout,K] (used as B^T)   -> LDS layout Bt[128][20]
// Block = 128 threads = 4 waves; block tile = 16(M) x 128(N); each wave owns
// two 16x16 tiles sharing one A fragment (2 WMMA per A load).
// Requires M%16==0, K%16==0, Nout%128==0 (M=50000, K in {128,256}, N in {256,1024}).
// ---------------------------------------------------------------------------
template <bool BT, bool ACC>
__global__ void gemm_wmma(const float* __restrict__ A, const float* __restrict__ B,
                          float* __restrict__ C, int M, int K, int Nout,
                          const float* __restrict__ bias0, const float* __restrict__ bias1) {
  __shared__ float As[16][20];    // 16x16 A tile, padded row (conflict-free col reads)
  __shared__ float Bsh[2560];     // BT: [128][20]; else: [16][132]

  const int m0   = blockIdx.x * 16;
  const int n0   = blockIdx.y * 128;
  const int tid  = threadIdx.x;
  const int lane = tid & 31;
  const int wave = tid >> 5;
  const int nsub = wave * 32;        // wave covers 32 N-columns (two 16-wide tiles)
  const int nn   = lane & 15;
  const int half = lane >> 4;        // K-pair / M-half selector per ISA layout

  v8f acc0 = {}, acc1 = {};

  for (int kb = 0; kb < K; kb += 16) {
    __syncthreads();   // previous iteration's LDS reads complete (WAR)

    // ---- stage A tile (16x16): thread t -> row t>>3, 2 floats (b64) ----
    {
      int r = tid >> 3, c = (tid & 7) * 2;
      async_ld_b64(&A[(size_t)(m0 + r) * K + kb + c], &As[r][c]);
    }
    // ---- stage B tile (16x128): 16 floats/thread as 4x b128 ----
    if (BT) {
      // column tid: 16 consecutive K floats -> Bt[tid][0..15]
      const float* gp = &B[(size_t)(n0 + tid) * K + kb];
      float* lp = &Bsh[tid * 20];
#pragma unroll
      for (int q = 0; q < 4; ++q) async_ld_b128(gp + 4 * q, lp + 4 * q);
    } else {
      int br = tid >> 3, bc = (tid & 7) * 16;
      const float* gp = &B[(size_t)(kb + br) * Nout + n0 + bc];
      float* lp = &Bsh[br * 132 + bc];
#pragma unroll
      for (int q = 0; q < 4; ++q) async_ld_b128(gp + 4 * q, lp + 4 * q);
    }
    wait_async0();     // our async copies landed in LDS
    __syncthreads();   // everyone's copies visible

#pragma unroll
    for (int kk = 0; kk < 16; kk += 4) {
      v2f a, b0, b1;
      // A 16x4 fragment: lane m = lane&15, K pair selected by lane[4]
      a.x = As[nn][kk + 2 * half];
      a.y = As[nn][kk + 2 * half + 1];
      // B 4x16 fragments for the wave's two N tiles
      if (BT) {
        int c0 = (nsub + nn) * 20, c1 = (nsub + 16 + nn) * 20;
        b0.x = Bsh[c0 + kk + 2 * half]; b0.y = Bsh[c0 + kk + 2 * half + 1];
        b1.x = Bsh[c1 + kk + 2 * half]; b1.y = Bsh[c1 + kk + 2 * half + 1];
      } else {
        int r0 = (kk + 2 * half) * 132, r1 = (kk + 2 * half + 1) * 132;
        b0.x = Bsh[r0 + nsub + nn];      b0.y = Bsh[r1 + nsub + nn];
        b1.x = Bsh[r0 + nsub + 16 + nn]; b1.y = Bsh[r1 + nsub + 16 + nn];
      }
      acc0 = __builtin_amdgcn_wmma_f32_16x16x4_f32(false, a, false, b0,
                                                   (short)0, acc0, false, false);
      acc1 = __builtin_amdgcn_wmma_f32_16x16x4_f32(false, a, false, b1,
                                                   (short)0, acc1, false, false);
    }
  }

  // ---- epilogue: C/D layout -> VGPR r holds row (r + 8*half), col = nn ----
  const int col0 = n0 + nsub + nn;
  const int col1 = col0 + 16;
  float badd0 = 0.0f, badd1 = 0.0f;
  if (bias0) { badd0 += bias0[col0]; badd1 += bias0[col1]; }
  if (bias1) { badd0 += bias1[col0]; badd1 += bias1[col1]; }

#pragma unroll
  for (int r = 0; r < 8; ++r) {
    size_t rowBase = (size_t)(m0 + r + 8 * half) * Nout;
    float v0 = acc0[r] + badd0;
    float v1 = acc1[r] + badd1;
    if (ACC) { v0 += C[rowBase + col0]; v1 += C[rowBase + col1]; }
    C[rowBase + col0] = v0;
    C[rowBase + col1] = v1;
  }
}

// ---------------------------------------------------------------------------
// per-node attention dot products: as[n]=h[n].a_s, ad[n]=h[n].a_d (wave/node)
// ---------------------------------------------------------------------------
__global__ void dots2_kernel(const float* __restrict__ h, const float* __restrict__ a_s,
                             const float* __restrict__ a_d, float* __restrict__ out_s,
                             float* __restrict__ out_d, int N) {
  int wid  = blockIdx.x * (blockDim.x >> 5) + (threadIdx.x >> 5);
  int lane = threadIdx.x & 31;
  if (wid >= N) return;
  const float* row = h + (size_t)wid * D_HID;
  float ss = 0.f, sd = 0.f;
  for (int c = lane; c < D_HID; c += 32) {
    float v = row[c];
    ss += v * a_s[c];
    sd += v * a_d[c];
  }
  for (int off = 16; off > 0; off >>= 1) {
    ss += __shfl_down(ss, off, 32);
    sd += __shfl_down(sd, off, 32);
  }
  if (lane == 0) { out_s[wid] = ss; out_d[wid] = sd; }
}

// ---------------------------------------------------------------------------
// edge pass 1: e = leaky_relu(as[src]+ad[dst], 0.2); segment max into emax[dst]
// indices >= E are self loops (src=dst=i-E)
// ---------------------------------------------------------------------------
__global__ void edge_max_kernel(const int* __restrict__ src, const int* __restrict__ dst,
                                const float* __restrict__ asv, const float* __restrict__ adv,
                                float* __restrict__ e_out, float* __restrict__ emax,
                                int E, int N) {
  int i = blockIdx.x * blockDim.x + threadIdx.x;
  if (i >= E + N) return;
  int s, d;
  if (i < E) { s = src[i]; d = dst[i]; } else { s = d = i - E; }
  float e = asv[s] + adv[d];
  e = (e > 0.f) ? e : 0.2f * e;
  e_out[i] = e;
  atomicMaxF(&emax[d], e);
}

// edge pass 2: w = exp(e - emax[dst]); denom[dst] += w; e_out <- w
__global__ void edge_w_kernel(const int* __restrict__ dst, float* __restrict__ e_out,
                              const float* __restrict__ emax, float* __restrict__ denom,
                              int E, int N) {
  int i = blockIdx.x * blockDim.x + threadIdx.x;
  if (i >= E + N) return;
  int d = (i < E) ? dst[i] : (i - E);
  float w = expf(e_out[i] - emax[d]);
  e_out[i] = w;
  atomicAdd(&denom[d], w);
}

// edge pass 3: acc[dst] += w * h[src]  (wave per edge, 8 channels/lane)
__global__ void edge_scatter_kernel(const int* __restrict__ src, const int* __restrict__ dst,
                                    const float* __restrict__ w, const float* __restrict__ h,
                                    float* __restrict__ acc, int E, int N) {
  int wid  = blockIdx.x * (blockDim.x >> 5) + (threadIdx.x >> 5);
  int lane = threadIdx.x & 31;
  if (wid >= E + N) return;
  int s, d;
  if (wid < E) { s = src[wid]; d = dst[wid]; } else { s = d = wid - E; }
  float ww = w[wid];
  const float* hr = h + (size_t)s * D_HID;
  float* ar = acc + (size_t)d * D_HID;
#pragma unroll
  for (int k = 0; k < 8; ++k) {
    int c = lane + 32 * k;
    atomicAdd(&ar[c], ww * hr[c]);
  }
}

// finalize: x = relu(acc/(denom+1e-16) + bias)
__global__ void gat_fin_kernel(const float* __restrict__ acc, const float* __restrict__ denom,
                               const float* __restrict__ bias, float* __restrict__ xout, int N) {
  size_t i = (size_t)blockIdx.x * blockDim.x + threadIdx.x;
  if (i >= (size_t)N * D_HID) return;
  size_t n = i >> 8;
  int c = (int)(i & (D_HID - 1));
  float v = acc[i] / (denom[n] + 1e-16f) + bias[c];
  xout[i] = fmaxf(v, 0.0f);
}

// ---------------------------------------------------------------------------
// LSTM pointwise cell. G: [N,1024] gate pre-acts (i,f,g,o). cprev nullable (=0).
// ---------------------------------------------------------------------------
__global__ void lstm_cell_kernel(const float* __restrict__ G, const float* __restrict__ cprev,
                                 float* __restrict__ h, float* __restrict__ c, int N) {
  size_t i = (size_t)blockIdx.x * blockDim.x + threadIdx.x;
  if (i >= (size_t)N * D_HID) return;
  size_t n = i >> 8;
  int j = (int)(i & (D_HID - 1));
  const float* g = G + n * (size_t)(4 * D_HID);
  float gi = g[j], gf = g[j + D_HID], gg = g[j + 2 * D_HID], go = g[j + 3 * D_HID];
  float cp = cprev ? cprev[i] : 0.0f;
  float cv = sigm(gf) * cp + sigm(gi) * tanhf(gg);
  c[i] = cv;
  h[i] = sigm(go) * tanhf(cv);
}

// score[t][n] = dot(hf[t][n], aw[0:256]) + dot(hb[t][n], aw[256:512])  (wave/node)
__global__ void score_kernel(const float* __restrict__ hf, const float* __restrict__ hb,
                             const float* __restrict__ aw, float* __restrict__ s, int N) {
  int wid  = blockIdx.x * (blockDim.x >> 5) + (threadIdx.x >> 5);
  int lane = threadIdx.x & 31;
  if (wid >= N) return;
  const float* pf = hf + (size_t)wid * D_HID;
  const float* pb = hb + (size_t)wid * D_HID;
  float a = 0.f;
  for (int c = lane; c < D_HID; c += 32) a += pf[c] * aw[c] + pb[c] * aw[c + D_HID];
  for (int off = 16; off > 0; off >>= 1) a += __shfl_down(a, off, 32);
  if (lane == 0) s[wid] = a;
}

// out[n] = softmax([s0,s1])[0]*x1[n] + softmax([s0,s1])[1]*x2[n]
__global__ void final_kernel(const float* __restrict__ s0, const float* __restrict__ s1,
                             const float* __restrict__ x1, const float* __restrict__ x2,
                             float* __restrict__ out, int N) {
  size_t i = (size_t)blockIdx.x * blockDim.x + threadIdx.x;
  if (i >= (size_t)N * D_HID) return;
  size_t n = i >> 8;
  float a = s0[n], b = s1[n];
  float m = fmaxf(a, b);
  float e0 = expf(a - m), e1 = expf(b - m);
  float inv = 1.0f / (e0 + e1);
  out[i] = (e0 * inv) * x1[i] + (e1 * inv) * x2[i];
}

// ---------------------------------------------------------------------------
// launch
// ---------------------------------------------------------------------------
extern "C" void kernel_launch(void* const* d_in, const int* in_sizes, int n_in,
                              void* d_out, int out_size, void* d_ws, size_t ws_size,
                              hipStream_t stream) {
  const float* x      = (const float*)d_in[0];
  const int*   eidx   = (const int*)d_in[1];
  const float* W1     = (const float*)d_in[2];
  const float* as1    = (const float*)d_in[3];
  const float* ad1    = (const float*)d_in[4];
  const float* b1     = (const float*)d_in[5];
  const float* W2     = (const float*)d_in[6];
  const float* as2    = (const float*)d_in[7];
  const float* ad2    = (const float*)d_in[8];
  const float* b2     = (const float*)d_in[9];
  const float* Wih_f  = (const float*)d_in[10];
  const float* Whh_f  = (const float*)d_in[11];
  const float* bih_f  = (const float*)d_in[12];
  const float* bhh_f  = (const float*)d_in[13];
  const float* Wih_b  = (const float*)d_in[14];
  const float* Whh_b  = (const float*)d_in[15];
  const float* bih_b  = (const float*)d_in[16];
  const float* bhh_b  = (const float*)d_in[17];
  const float* attn_w = (const float*)d_in[18];

  const int N   = in_sizes[0] / D_IN;   // 50000 (divisible by 16)
  const int E   = in_sizes[1] / 2;      // 800000
  const int TOT = E + N;
  const int* srcp = eidx;
  const int* dstp = eidx + E;

  const size_t ND = (size_t)N * D_HID;

  // ---- workspace layout (floats), with GAT/LSTM scratch aliased in R ----
  float* ws  = (float*)d_ws;
  float* x1  = ws;
  float* x2  = x1 + ND;
  float* hf0 = x2 + ND;
  float* hf1 = hf0 + ND;
  float* hb0 = hf1 + ND;
  float* hb1 = hb0 + ND;
  float* s0  = hb1 + ND;   // aliased: emax during GAT
  float* s1  = s0 + N;     // aliased: denom during GAT
  float* asv = s1 + N;
  float* adv = asv + N;
  float* R   = adv + N;    // scratch region, size 6*ND
  // GAT-phase view of R
  float* hbuf = R;             // [N,256] pre-activation h
  float* accb = R + ND;        // [N,256] aggregation accumulator
  float* eedg = R + 2 * ND;    // [E+N] edge logits -> weights
  // LSTM-phase view of R
  float* G  = R;               // [N,1024] gates
  float* cA = R + 4 * ND;      // cell state ping
  float* cB = cA + ND;         // cell state pong

  const dim3 blk128(128), blk256(256);
  const unsigned gND   = (unsigned)((ND + 255) / 256);
  const unsigned gN    = (unsigned)((N + 255) / 256);
  const unsigned gTOT  = (unsigned)((TOT + 255) / 256);
  const unsigned gWN   = (unsigned)((N + 7) / 8);    // wave-per-node, 8 waves/block
  const unsigned gWTOT = (unsigned)((TOT + 7) / 8);  // wave-per-edge

  // ---- one GAT layer ----
  auto run_gat = [&](const float* xin, int Kin, const float* W, const float* av_s,
                     const float* av_d, const float* bias, float* xout) {
    dim3 gg(N / 16, D_HID / 128);
    gemm_wmma<false, false><<<gg, blk128, 0, stream>>>(xin, W, hbuf, N, Kin, D_HID,
                                                       nullptr, nullptr);
    dots2_kernel<<<gWN, blk256, 0, stream>>>(hbuf, av_s, av_d, asv, adv, N);
    fill_kernel<<<gN, blk256, 0, stream>>>(s0, -1e30f, N);          // emax
    fill_kernel<<<gN, blk256, 0, stream>>>(s1, 0.0f, N);            // denom
    fill_kernel<<<gND, blk256, 0, stream>>>(accb, 0.0f, (int)ND);   // acc
    edge_max_kernel<<<gTOT, blk256, 0, stream>>>(srcp, dstp, asv, adv, eedg, s0, E, N);
    edge_w_kernel<<<gTOT, blk256, 0, stream>>>(dstp, eedg, s0, s1, E, N);
    edge_scatter_kernel<<<gWTOT, blk256, 0, stream>>>(srcp, dstp, eedg, hbuf, accb, E, N);
    gat_fin_kernel<<<gND, blk256, 0, stream>>>(accb, s1, bias, xout, N);
  };

  run_gat(x, D_IN, W1, as1, ad1, b1, x1);
  run_gat(x1, D_HID, W2, as2, ad2, b2, x2);

  // ---- LSTM gate GEMMs: G (+)= A @ W^T (+bih+bhh) ----
  const dim3 ggL(N / 16, (4 * D_HID) / 128);
  auto gemmT = [&](const float* A, const float* Wt, const float* bb0, const float* bb1) {
    gemm_wmma<true, false><<<ggL, blk128, 0, stream>>>(A, Wt, G, N, D_HID, 4 * D_HID,
                                                       bb0, bb1);
  };
  auto gemmTacc = [&](const float* A, const float* Wt) {
    gemm_wmma<true, true><<<ggL, blk128, 0, stream>>>(A, Wt, G, N, D_HID, 4 * D_HID,
                                                      nullptr, nullptr);
  };

  // forward direction over [x1, x2]
  gemmT(x1, Wih_f, bih_f, bhh_f);
  lstm_cell_kernel<<<gND, blk256, 0, stream>>>(G, nullptr, hf0, cA, N);
  gemmT(x2, Wih_f, bih_f, bhh_f);
  gemmTacc(hf0, Whh_f);
  lstm_cell_kernel<<<gND, blk256, 0, stream>>>(G, cA, hf1, cB, N);

  // backward direction over [x2, x1]; hb1 = first step, hb0 = second step
  gemmT(x2, Wih_b, bih_b, bhh_b);
  lstm_cell_kernel<<<gND, blk256, 0, stream>>>(G, nullptr, hb1, cA, N);
  gemmT(x1, Wih_b, bih_b, bhh_b);
  gemmTacc(hb1, Whh_b);
  lstm_cell_kernel<<<gND, blk256, 0, stream>>>(G, cA, hb0, cB, N);

  // ---- layer attention + blend ----
  score_kernel<<<gWN, blk256, 0, stream>>>(hf0, hb0, attn_w, s0, N);
  score_kernel<<<gWN, blk256, 0, stream>>>(hf1, hb1, attn_w, s1, N);
  final_kernel<<<gND, blk256, 0, stream>>>(s0, s1, x1, x2, (float*)d_out, N);
}